// SoftHeatMap_16183436771723
// MI455X (gfx1250) — compile-verified
//
#include <hip/hip_runtime.h>
#include <hip/hip_bf16.h>

namespace {

constexpr int W = 4096;
constexpr int H = 4096;

#if __has_builtin(__builtin_amdgcn_global_load_async_to_lds_b128) && \
    __has_builtin(__builtin_amdgcn_s_wait_asynccnt)
#define SHM_HAVE_ASYNC_LDS 1
typedef int shm_v4i __attribute__((ext_vector_type(4)));
typedef __attribute__((address_space(1))) shm_v4i shm_gv4i;  // global
typedef __attribute__((address_space(3))) shm_v4i shm_lv4i;  // LDS
#endif

// ---------------------------------------------------------------------------
// Kernel 1: bulk zero-fill with 128-bit stores (bandwidth floor of the op).
// ---------------------------------------------------------------------------
__global__ __launch_bounds__(256) void shm_fill_zero(float4* __restrict__ p,
                                                     unsigned long long n4) {
  unsigned long long i =
      (unsigned long long)blockIdx.x * blockDim.x + threadIdx.x;
  if (i < n4) p[i] = make_float4(0.f, 0.f, 0.f, 0.f);
}

// ---------------------------------------------------------------------------
// Kernel 2: one thread per box.
//  - stage this block's 256 boxes into LDS via gfx1250 async LDS-DMA
//  - scatter-max the 3x3 gaussian stamp with integer atomicMax on float bits
//    (all values >= 0, so int order == float order; order-independent)
//  - claim the sizemap pixel deterministically: highest box index wins
// ---------------------------------------------------------------------------
__global__ __launch_bounds__(256) void shm_box_pass1(
    const float4* __restrict__ boxes, const float* __restrict__ mount,
    int* __restrict__ heat_bits, unsigned* __restrict__ winner,
    float* __restrict__ size0, float* __restrict__ size1, int n) {
  const int b = blockIdx.x * 256 + threadIdx.x;

  float4 bx;
  float m[9];
#ifdef SHM_HAVE_ASYNC_LDS
  __shared__ float4 sbox[256];
  if (b < n) {
    // 16B per lane, global -> LDS, tracked by ASYNCcnt (no VGPR data path)
    __builtin_amdgcn_global_load_async_to_lds_b128(
        (shm_gv4i*)(boxes + b),
        (shm_lv4i*)&sbox[threadIdx.x],
        /*offset=*/0, /*cpol=*/0);
  }
  // Overlap the 3x3 stamp load with the in-flight async copy.
#pragma unroll
  for (int k = 0; k < 9; ++k) m[k] = mount[k];
  __builtin_amdgcn_s_wait_asynccnt(0);
  if (b >= n) return;
  bx = sbox[threadIdx.x];  // per-lane slot: no cross-wave barrier required
#else
#pragma unroll
  for (int k = 0; k < 9; ++k) m[k] = mount[k];
  if (b >= n) return;
  bx = boxes[b];
#endif

  int cx = (int)(bx.x * (float)W);  // trunc == floor (inputs >= 0)
  int cy = (int)(bx.y * (float)H);
  cx = min(max(cx, 0), W - 1);
  cy = min(max(cy, 0), H - 1);
  const unsigned p = (unsigned)cy * W + (unsigned)cx;

  if (winner) {
    atomicMax(&winner[p], (unsigned)(b + 1));  // deterministic claim
  } else {
    // fallback (tiny ws): racy "one write wins", matches jax's unspecified dup
    size0[p] = bx.z;
    size1[p] = bx.w;
  }

  // Stamp only when the full 3x3 window fits (reference semantics).
  if (cx >= 1 && cx <= W - 2 && cy >= 1 && cy <= H - 2) {
#pragma unroll
    for (int dy = -1; dy <= 1; ++dy) {
#pragma unroll
      for (int dx = -1; dx <= 1; ++dx) {
        const unsigned q = (unsigned)(cy + dy) * W + (unsigned)(cx + dx);
        // All heat values >= 0: signed-int max == float max. Resolves in L2
        // (192MB output == L2 capacity).
        atomicMax(&heat_bits[q], __float_as_int(m[(dy + 1) * 3 + (dx + 1)]));
      }
    }
  }
}

// ---------------------------------------------------------------------------
// Kernel 3: winners write their (w,h) — race-free, deterministic.
// ---------------------------------------------------------------------------
__global__ __launch_bounds__(256) void shm_box_pass2(
    const float4* __restrict__ boxes, const unsigned* __restrict__ winner,
    float* __restrict__ size0, float* __restrict__ size1, int n) {
  const int b = blockIdx.x * 256 + threadIdx.x;
  if (b >= n) return;
  __builtin_prefetch(boxes + b, 0, 0);  // global_prefetch_b8
  const float4 bx = boxes[b];
  int cx = (int)(bx.x * (float)W);
  int cy = (int)(bx.y * (float)H);
  cx = min(max(cx, 0), W - 1);
  cy = min(max(cy, 0), H - 1);
  const unsigned p = (unsigned)cy * W + (unsigned)cx;
  if (winner[p] == (unsigned)(b + 1)) {
    size0[p] = bx.z;
    size1[p] = bx.w;
  }
}

}  // namespace

extern "C" void kernel_launch(void* const* d_in, const int* in_sizes, int n_in,
                              void* d_out, int out_size, void* d_ws,
                              size_t ws_size, hipStream_t stream) {
  const float4* boxes = (const float4*)d_in[0];   // [N,4] cx,cy,w,h
  const float* mount = (const float*)d_in[1];     // 3x3 gaussian
  const int nboxes = in_sizes[0] / 4;

  float* out = (float*)d_out;
  float* heat = out;                               // [H*W]
  float* size0 = out + (size_t)H * W;              // sizemap plane 0
  float* size1 = out + 2ull * (size_t)H * W;       // sizemap plane 1

  unsigned* winner = nullptr;
  if (ws_size >= (size_t)H * W * sizeof(unsigned))
    winner = (unsigned*)d_ws;

  // Zero the 3 output planes (192 MB) with B128 stores.
  {
    const unsigned long long n4 = (3ull * H * W) / 4;
    shm_fill_zero<<<(unsigned)((n4 + 255) / 256), 256, 0, stream>>>(
        (float4*)out, n4);
  }
  // Zero the winner plane.
  if (winner) {
    const unsigned long long n4 = ((unsigned long long)H * W) / 4;
    shm_fill_zero<<<(unsigned)((n4 + 255) / 256), 256, 0, stream>>>(
        (float4*)winner, n4);
  }

  const unsigned bgrid = (unsigned)((nboxes + 255) / 256);
  shm_box_pass1<<<bgrid, 256, 0, stream>>>(boxes, mount, (int*)heat, winner,
                                           size0, size1, nboxes);
  if (winner) {
    shm_box_pass2<<<bgrid, 256, 0, stream>>>(boxes, winner, size0, size1,
                                             nboxes);
  }
}